// SSMBlock_88304527606383
// MI455X (gfx1250) — compile-verified
//
#include <hip/hip_runtime.h>
#include <math.h>

typedef __attribute__((ext_vector_type(2))) float v2f;
typedef __attribute__((ext_vector_type(8))) float v8f;

#define D_MODEL 2048
#define D_STATE 16
#define SEQ     4096
#define BATCH   4
#define NTOK    (BATCH * SEQ)
#define LN_EPS  1e-5f

// -------------------------------------------------------------------------
// Phase 1: fused LayerNorm stats + v = xn @ B^T  (skinny GEMM via fp32 WMMA)
// Grid: NTOK/16 blocks of 256 threads (8 waves). Each block owns 16 tokens.
// Wave w reduces LN stats for rows 2w,2w+1, then handles K-slice [w*256,(w+1)*256)
// of the 2048-deep dot products; partial 16x16 accumulators reduced via LDS.
// Outputs: mu[tok], rstd[tok], v[tok][16].  xn is NOT materialized.
// -------------------------------------------------------------------------
__global__ __launch_bounds__(256) void ssm_phase1(
    const float* __restrict__ x, const float* __restrict__ Bm,
    const float* __restrict__ gamma, const float* __restrict__ beta,
    float* __restrict__ mu_g, float* __restrict__ rstd_g,
    float* __restrict__ v_g)
{
    __shared__ float mu_s[16];
    __shared__ float rstd_s[16];
    __shared__ float partial[8 * 256];

    const int  tid   = threadIdx.x;
    const int  wave  = tid >> 5;
    const int  lane  = tid & 31;
    const long token0 = (long)blockIdx.x * 16;

    // ---- LayerNorm statistics (wave w -> rows 2w, 2w+1), float4 loads ----
    for (int rr = 0; rr < 2; ++rr) {
        const int row = 2 * wave + rr;
        const float4* xr = (const float4*)(x + (token0 + row) * D_MODEL);
        float s = 0.f, s2 = 0.f;
        #pragma unroll
        for (int it = 0; it < D_MODEL / 128; ++it) {
            float4 xv = xr[it * 32 + lane];
            s  += xv.x + xv.y + xv.z + xv.w;
            s2 += xv.x * xv.x + xv.y * xv.y + xv.z * xv.z + xv.w * xv.w;
        }
        #pragma unroll
        for (int off = 16; off; off >>= 1) {
            s  += __shfl_xor(s,  off, 32);
            s2 += __shfl_xor(s2, off, 32);
        }
        if (lane == 0) {
            const float mu   = s * (1.f / D_MODEL);
            const float var  = s2 * (1.f / D_MODEL) - mu * mu;
            const float rstd = rsqrtf(var + LN_EPS);
            mu_s[row]   = mu;   rstd_s[row]   = rstd;
            mu_g[token0 + row] = mu;
            rstd_g[token0 + row] = rstd;
        }
    }
    __syncthreads();

    // ---- WMMA K-slice: A-frag = on-the-fly normalized x, B-frag = B^T ----
    // V_WMMA_F32_16X16X4_F32 A layout: lanes 0-15 -> M=lane, K=k0,k0+1;
    //                                  lanes 16-31 -> M=lane-16, K=k0+2,k0+3.
    const int arow  = lane & 15;
    const int khalf = (lane >> 4) * 2;
    const float mu   = mu_s[arow];
    const float rstd = rstd_s[arow];
    const float* xrow = x  + (token0 + arow) * D_MODEL;
    const float* brow = Bm + (long)(lane & 15) * D_MODEL;   // n = lane&15 (B^T column)

    v8f acc = {};
    const int k0base = wave * 256;
    for (int kk = 0; kk < 256; kk += 4) {
        const int kb = k0base + kk + khalf;
        const float2 xv = *(const float2*)(xrow + kb);
        const float2 gv = *(const float2*)(gamma + kb);
        const float2 bv = *(const float2*)(beta + kb);
        v2f a, b2;
        a.x = (xv.x - mu) * rstd * gv.x + bv.x;
        a.y = (xv.y - mu) * rstd * gv.y + bv.y;
        const float2 bm = *(const float2*)(brow + kb);
        b2.x = bm.x; b2.y = bm.y;
        acc = __builtin_amdgcn_wmma_f32_16x16x4_f32(
            false, a, false, b2, (short)0, acc, false, false);
    }

    // ---- cross-wave reduction of partial 16x16 accumulators ----
    #pragma unroll
    for (int r = 0; r < 8; ++r)
        partial[wave * 256 + r * 32 + lane] = acc[r];
    __syncthreads();

    float sum = 0.f;
    #pragma unroll
    for (int w = 0; w < 8; ++w)
        sum += partial[w * 256 + tid];

    // D layout: elem(tid) -> r=tid>>5, lane=tid&31, row=r+((lane>>4)<<3), col=lane&15
    const int r   = tid >> 5;
    const int l2  = tid & 31;
    const int row = r + ((l2 >> 4) << 3);
    const int col = l2 & 15;
    v_g[(token0 + row) * D_STATE + col] = sum;
}

// -------------------------------------------------------------------------
// Phase 2: serial tanh recurrence. One wave per batch element.
// Lane j owns state j; h broadcast via wave shuffles (wave32).
// -------------------------------------------------------------------------
__global__ __launch_bounds__(32) void ssm_phase2(
    const float* __restrict__ Am, const float* __restrict__ v_g,
    float* __restrict__ H)
{
    const int b    = blockIdx.x;
    const int lane = threadIdx.x;
    const int j    = lane & 15;          // lanes 16-31 mirror 0-15 (harmless)

    float a[16];
    #pragma unroll
    for (int i = 0; i < 16; ++i) a[i] = Am[i * 16 + j];   // column j of A

    const float* vb = v_g + (long)b * SEQ * D_STATE;
    float*       Hb = H   + (long)b * SEQ * D_STATE;

    float h = 0.f;
    for (int t = 0; t < SEQ; ++t) {
        // tree-partition the 16-term dot product to shorten the FMA chain
        float p0 = vb[t * D_STATE + j];
        float p1 = 0.f, p2 = 0.f, p3 = 0.f;
        #pragma unroll
        for (int i = 0; i < 4; ++i) {
            p0 += __shfl(h, i,      32) * a[i];
            p1 += __shfl(h, i + 4,  32) * a[i + 4];
            p2 += __shfl(h, i + 8,  32) * a[i + 8];
            p3 += __shfl(h, i + 12, 32) * a[i + 12];
        }
        h = tanhf((p0 + p1) + (p2 + p3));
        if (lane < 16) Hb[t * D_STATE + lane] = h;
    }
}

// -------------------------------------------------------------------------
// Phase 3: out = H @ C + (Dp+1) * xn, xn recomputed from x + (mu,rstd).
// Grid: (D_MODEL/128, NTOK/16); block 256 = 8 waves, each wave one 16-col tile.
// K=16 -> 4 fp32 WMMAs per tile; H tile staged in LDS (shared by all 8 waves).
// -------------------------------------------------------------------------
__global__ __launch_bounds__(256) void ssm_phase3(
    const float* __restrict__ x,  const float* __restrict__ Cm,
    const float* __restrict__ Dp, const float* __restrict__ gamma,
    const float* __restrict__ beta, const float* __restrict__ mu_g,
    const float* __restrict__ rstd_g, const float* __restrict__ H,
    float* __restrict__ out)
{
    __shared__ float h_s[16 * 16];
    __shared__ float mu_s[16];
    __shared__ float rstd_s[16];

    const int  tid    = threadIdx.x;
    const int  wave   = tid >> 5;
    const int  lane   = tid & 31;
    const long token0 = (long)blockIdx.y * 16;
    const int  ncol0  = blockIdx.x * 128 + wave * 16;

    h_s[tid] = H[token0 * D_STATE + tid];          // 16 tokens x 16 states
    if (tid < 16) {
        mu_s[tid]   = mu_g[token0 + tid];
        rstd_s[tid] = rstd_g[token0 + tid];
    }
    __syncthreads();

    const int arow  = lane & 15;
    const int khalf = (lane >> 4) * 2;
    const int ncol  = ncol0 + (lane & 15);

    v8f acc = {};
    #pragma unroll
    for (int k0 = 0; k0 < D_STATE; k0 += 4) {
        const int kb = k0 + khalf;
        v2f a, bb;
        a.x = h_s[arow * 16 + kb];
        a.y = h_s[arow * 16 + kb + 1];
        bb.x = Cm[(long)kb * D_MODEL + ncol];        // C used untransposed: B[k][n]=C[k][n]
        bb.y = Cm[(long)(kb + 1) * D_MODEL + ncol];
        acc = __builtin_amdgcn_wmma_f32_16x16x4_f32(
            false, a, false, bb, (short)0, acc, false, false);
    }

    const float g   = gamma[ncol];
    const float be  = beta[ncol];
    const float dp1 = Dp[ncol] + 1.f;
    #pragma unroll
    for (int r = 0; r < 8; ++r) {
        const int  row = r + ((lane >> 4) << 3);
        const long idx = (token0 + row) * D_MODEL + ncol;
        const float xn = (x[idx] - mu_s[row]) * rstd_s[row] * g + be;
        out[idx] = acc[r] + dp1 * xn;
    }
}

// -------------------------------------------------------------------------
extern "C" void kernel_launch(void* const* d_in, const int* in_sizes, int n_in,
                              void* d_out, int out_size, void* d_ws, size_t ws_size,
                              hipStream_t stream) {
    const float* x     = (const float*)d_in[0];
    const float* Am    = (const float*)d_in[1];
    const float* Bm    = (const float*)d_in[2];
    const float* Cm    = (const float*)d_in[3];
    const float* Dp    = (const float*)d_in[4];
    const float* gamma = (const float*)d_in[5];
    const float* beta  = (const float*)d_in[6];
    float* out = (float*)d_out;

    // workspace carve-out (~2.25 MB total)
    float* mu   = (float*)d_ws;            // NTOK
    float* rstd = mu + NTOK;               // NTOK
    float* v    = rstd + NTOK;             // NTOK * D_STATE
    float* H    = v + (long)NTOK * D_STATE;// NTOK * D_STATE

    ssm_phase1<<<NTOK / 16, 256, 0, stream>>>(x, Bm, gamma, beta, mu, rstd, v);
    ssm_phase2<<<BATCH, 32, 0, stream>>>(Am, v, H);
    ssm_phase3<<<dim3(D_MODEL / 128, NTOK / 16), 256, 0, stream>>>(
        x, Cm, Dp, gamma, beta, mu, rstd, H, out);
}